// UncertaintyWeightedLoss_42090679501421
// MI455X (gfx1250) — compile-verified
//
#include <hip/hip_runtime.h>
#include <hip/hip_bf16.h>

// ---------------------------------------------------------------------------
// UncertaintyWeightedLoss for MI455X (gfx1250)
// Memory-bound segmented softmax-CE: stream logits HBM->LDS once via CDNA5
// async-to-LDS loads, reduce in LDS, use V_WMMA_F32_16X16X4_F32 (ones-matrix
// trick) for exact f32 wave-sum reductions.
// ---------------------------------------------------------------------------

typedef __attribute__((ext_vector_type(2))) float v2f;
typedef __attribute__((ext_vector_type(8))) float v8f;

#define SEG_BLOCK 128
#define FIN_BLOCK 1024
#define CHUNK 2048   // floats of LDS staging per chunk (8 KB)

// ---- LDS address helper: generic ptr -> 32-bit LDS byte offset -------------
static __device__ __forceinline__ unsigned ldsAddr(const void* p) {
    return (unsigned)(unsigned long long)
        (__attribute__((address_space(3))) const void*)p;
}

// ---- async wait ------------------------------------------------------------
static __device__ __forceinline__ void waitAsync0() {
#if __has_builtin(__builtin_amdgcn_s_wait_asynccnt)
    __builtin_amdgcn_s_wait_asynccnt(0);
#else
    asm volatile("s_wait_asynccnt 0" ::: "memory");
#endif
}

// ---- async global -> LDS loads --------------------------------------------
static __device__ __forceinline__ void asyncLoadB128(const float* g, const float* l) {
    unsigned la = ldsAddr(l);
    asm volatile("global_load_async_to_lds_b128 %0, %1, off"
                 :: "v"(la), "v"(g) : "memory");
}
static __device__ __forceinline__ void asyncLoadB32(const float* g, const float* l) {
    unsigned la = ldsAddr(l);
    asm volatile("global_load_async_to_lds_b32 %0, %1, off"
                 :: "v"(la), "v"(g) : "memory");
}

// ---- exact f32 wave(32) sum via V_WMMA_F32_16X16X4_F32 ---------------------
// A: lane m (m<16): VGPR0=A[m][0], VGPR1=A[m][1]; lane 16+m: VGPR0=A[m][2],
// VGPR1=A[m][3]. Put v in VGPR0, 0 in VGPR1 -> A[m][0]=v_m, A[m][2]=v_{m+16}.
// B = ones(4x16) -> D[m][n] = v_m + v_{m+16} for all n.
// D: VGPR r, lanes 0-15 hold D[r][n]; lanes 16-31 hold D[8+r][n]. So per-lane
// sum of the 8 D components is rowsums(0..7) (lanes<16) or rowsums(8..15)
// (lanes>=16); one xor-16 shuffle completes the exact f32 wave sum.
static __device__ __forceinline__ float waveSum(float v) {
#if __has_builtin(__builtin_amdgcn_wmma_f32_16x16x4_f32)
    v2f a; a[0] = v;   a[1] = 0.0f;
    v2f b; b[0] = 1.0f; b[1] = 1.0f;
    v8f c = {};
    c = __builtin_amdgcn_wmma_f32_16x16x4_f32(
        /*neg_a=*/false, a, /*neg_b=*/false, b,
        /*c_mod=*/(short)0, c, /*reuse_a=*/false, /*reuse_b=*/false);
    float s = c[0] + c[1] + c[2] + c[3] + c[4] + c[5] + c[6] + c[7];
    s += __shfl_xor(s, 16, 32);
    return s;
#else
    for (int o = 16; o > 0; o >>= 1) v += __shfl_xor(v, o, 32);
    return v;
#endif
}

// ---- block reductions (all threads get the result) -------------------------
template<int BLOCK>
static __device__ __forceinline__ float blockSum(float v, float* sc) {
    v = waveSum(v);
    int wid = threadIdx.x >> 5;
    __syncthreads();
    if ((threadIdx.x & 31) == 0) sc[wid] = v;
    __syncthreads();
    float r = 0.0f;
#pragma unroll
    for (int w = 0; w < BLOCK / 32; ++w) r += sc[w];
    return r;
}

template<int BLOCK>
static __device__ __forceinline__ float blockMax(float v, float* sc) {
    for (int o = 16; o > 0; o >>= 1) v = fmaxf(v, __shfl_xor(v, o, 32));
    int wid = threadIdx.x >> 5;
    __syncthreads();
    if ((threadIdx.x & 31) == 0) sc[wid] = v;
    __syncthreads();
    float r = sc[0];
#pragma unroll
    for (int w = 1; w < BLOCK / 32; ++w) r = fmaxf(r, sc[w]);
    return r;
}

template<int BLOCK>
static __device__ __forceinline__ int blockMinI(int v, int* sc) {
    for (int o = 16; o > 0; o >>= 1) v = min(v, __shfl_xor(v, o, 32));
    int wid = threadIdx.x >> 5;
    __syncthreads();
    if ((threadIdx.x & 31) == 0) sc[wid] = v;
    __syncthreads();
    int r = sc[0];
#pragma unroll
    for (int w = 1; w < BLOCK / 32; ++w) r = min(r, sc[w]);
    return r;
}

// ---------------------------------------------------------------------------
// Kernel 1: exclusive prefix-sum of counts -> per-segment offsets.
// blockIdx.x == 0 -> sys, == 1 -> bar. B=16K: trivial single-block scan.
// ---------------------------------------------------------------------------
__global__ __launch_bounds__(256)
void scan_kernel(const int* __restrict__ c0, const int* __restrict__ c1,
                 int* __restrict__ o0, int* __restrict__ o1, int Bn) {
    const int* c = blockIdx.x ? c1 : c0;
    int* o       = blockIdx.x ? o1 : o0;
    __shared__ int tsum[256];
    __shared__ int tpre[256];
    int chunk = (Bn + 255) / 256;
    int s0 = threadIdx.x * chunk;
    int e0 = min(s0 + chunk, Bn);
    int s = 0;
    for (int i = s0; i < e0; ++i) s += c[i];
    tsum[threadIdx.x] = s;
    __syncthreads();
    if (threadIdx.x == 0) {
        int acc = 0;
        for (int t = 0; t < 256; ++t) { tpre[t] = acc; acc += tsum[t]; }
    }
    __syncthreads();
    int run = tpre[threadIdx.x];
    for (int i = s0; i < e0; ++i) { o[i] = run; run += c[i]; }
}

// ---------------------------------------------------------------------------
// Kernel 2: per-segment online-softmax CE. One block per (segment, tensor).
// Streams the segment through LDS once (async loads), computes max, first
// argmax, sum of exp, merges chunks online. Thread 0 writes masked CE + flags.
// ---------------------------------------------------------------------------
__global__ __launch_bounds__(SEG_BLOCK)
void seg_ce_kernel(const float* __restrict__ sys_logits,
                   const float* __restrict__ bar_logits,
                   const int* __restrict__ off_sys, const int* __restrict__ off_bar,
                   const int* __restrict__ cnt_sys, const int* __restrict__ cnt_bar,
                   const int* __restrict__ gt_sys,  const int* __restrict__ gt_bar,
                   const unsigned char* __restrict__ valid,
                   float* __restrict__ ce_sys, float* __restrict__ ce_bar,
                   int* __restrict__ fl_sys,  int* __restrict__ fl_bar, int Bn) {
    __shared__ float smem[CHUNK];
    __shared__ float sred[SEG_BLOCK / 32];
    __shared__ int   sredi[SEG_BLOCK / 32];

    int bid   = blockIdx.x;
    int which = (bid >= Bn) ? 1 : 0;
    int seg   = which ? (bid - Bn) : bid;
    const float* logits = which ? bar_logits : sys_logits;
    int count = which ? cnt_bar[seg] : cnt_sys[seg];
    int off   = which ? off_bar[seg] : off_sys[seg];
    int gt    = which ? gt_bar[seg]  : gt_sys[seg];
    int tid   = threadIdx.x;

    const float NEG_INF = -__builtin_huge_valf();
    float run_max = NEG_INF;
    float run_sum = 0.0f;
    int   run_arg = 0x7fffffff;

    for (int base = 0; base < count; base += CHUNK) {
        int n = min(CHUNK, count - base);
        const float* src = logits + off + base;
        int nb = n & ~3;
        // bulk: 16B async copies straight into LDS (per-lane addresses)
        for (int i = tid * 4; i < nb; i += SEG_BLOCK * 4)
            asyncLoadB128(src + i, &smem[i]);
        // tail (<4 elements)
        if (tid < (n - nb))
            asyncLoadB32(src + nb + tid, &smem[nb + tid]);
        waitAsync0();
        __syncthreads();

        // chunk max
        float lm = NEG_INF;
        for (int i = tid; i < n; i += SEG_BLOCK) lm = fmaxf(lm, smem[i]);
        float cmax = blockMax<SEG_BLOCK>(lm, sred);

        // chunk first-occurrence argmax (local index within segment)
        int la = 0x7fffffff;
        for (int i = tid; i < n; i += SEG_BLOCK)
            if (smem[i] == cmax) la = min(la, base + i);
        int carg = blockMinI<SEG_BLOCK>(la, sredi);

        // chunk sum of exp against the merged max (exact f32 via WMMA reduce)
        float nmax = fmaxf(run_max, cmax);
        float ls = 0.0f;
        for (int i = tid; i < n; i += SEG_BLOCK) ls += expf(smem[i] - nmax);
        float csum = blockSum<SEG_BLOCK>(ls, sred);

        run_sum = run_sum * expf(run_max - nmax) + csum;
        if (cmax > run_max) run_arg = carg;   // tie keeps earlier chunk (first occurrence)
        run_max = nmax;
        __syncthreads();                       // smem reuse fence for next chunk
    }

    if (tid == 0) {
        bool in_range = (gt >= 0) && (gt < count);
        bool vm = (count > 0) && in_range && (valid[seg] != 0);
        float ce = 0.0f;
        int corr = 0;
        if (vm) {
            float lse = logf(run_sum) + run_max;
            int pos = off + min(max(gt, 0), max(count - 1, 0));
            ce = lse - logits[pos];
            corr = (run_arg == gt) ? 1 : 0;
        }
        int fl = (vm ? 1 : 0) | (corr << 1);
        if (which) { ce_bar[seg] = ce; fl_bar[seg] = fl; }
        else       { ce_sys[seg] = ce; fl_sys[seg] = fl; }
    }
}

// ---------------------------------------------------------------------------
// Kernel 3: final reduction over B segments + scalar combine -> 9 outputs.
// ---------------------------------------------------------------------------
__global__ __launch_bounds__(FIN_BLOCK)
void finalize_kernel(const float* __restrict__ ce_s, const int* __restrict__ fl_s,
                     const float* __restrict__ ce_b, const int* __restrict__ fl_b,
                     const float* __restrict__ note_pos,
                     const float* __restrict__ gt_note,
                     const unsigned char* __restrict__ valid,
                     const float* __restrict__ lv_sys,
                     const float* __restrict__ lv_bar,
                     const float* __restrict__ lv_note,
                     float* __restrict__ out, int Bn) {
    __shared__ float sred[FIN_BLOCK / 32];

    float s_ces = 0.0f, s_ceb = 0.0f, s_sq = 0.0f;
    float nv_s = 0.0f, nv_b = 0.0f, nc_s = 0.0f, nc_b = 0.0f, n_note = 0.0f;

    for (int i = threadIdx.x; i < Bn; i += FIN_BLOCK) {
        s_ces += ce_s[i];
        s_ceb += ce_b[i];
        int fs = fl_s[i], fb = fl_b[i];
        nv_s += (float)(fs & 1);
        nc_s += (float)((fs >> 1) & 1);
        nv_b += (float)(fb & 1);
        nc_b += (float)((fb >> 1) & 1);
        float v = valid[i] ? 1.0f : 0.0f;
        n_note += v;
        float d = note_pos[i] - gt_note[i];
        s_sq += v * d * d;
    }

    s_ces  = blockSum<FIN_BLOCK>(s_ces, sred);
    s_ceb  = blockSum<FIN_BLOCK>(s_ceb, sred);
    s_sq   = blockSum<FIN_BLOCK>(s_sq, sred);
    nv_s   = blockSum<FIN_BLOCK>(nv_s, sred);
    nv_b   = blockSum<FIN_BLOCK>(nv_b, sred);
    nc_s   = blockSum<FIN_BLOCK>(nc_s, sred);
    nc_b   = blockSum<FIN_BLOCK>(nc_b, sred);
    n_note = blockSum<FIN_BLOCK>(n_note, sred);

    if (threadIdx.x == 0) {
        float sys_loss  = s_ces / fmaxf(nv_s, 1.0f);
        float bar_loss  = s_ceb / fmaxf(nv_b, 1.0f);
        float note_loss = (n_note > 0.0f) ? (s_sq / fmaxf(n_note, 1.0f)) : 0.0f;
        float ls = lv_sys[0], lb = lv_bar[0], ln = lv_note[0];
        float ps = expf(-ls), pb = expf(-lb), pn = expf(-ln);
        float loss = 0.5f * (ps * sys_loss + ls + pb * bar_loss + lb
                             + pn * note_loss + ln);
        out[0] = loss;
        out[1] = sys_loss;
        out[2] = bar_loss;
        out[3] = note_loss;
        out[4] = nc_s / fmaxf(nv_s, 1.0f);
        out[5] = nc_b / fmaxf(nv_b, 1.0f);
        out[6] = ps;
        out[7] = pb;
        out[8] = pn;
    }
}

// ---------------------------------------------------------------------------
extern "C" void kernel_launch(void* const* d_in, const int* in_sizes, int n_in,
                              void* d_out, int out_size, void* d_ws, size_t ws_size,
                              hipStream_t stream) {
    const float* sys_logits        = (const float*)d_in[0];
    const int*   sys_counts        = (const int*)d_in[1];
    const float* bar_logits        = (const float*)d_in[2];
    const int*   bar_counts        = (const int*)d_in[3];
    const float* note_positions    = (const float*)d_in[4];
    const int*   gt_system_idx     = (const int*)d_in[5];
    const int*   gt_bar_in_sys     = (const int*)d_in[6];
    const float* gt_note_position  = (const float*)d_in[7];
    const unsigned char* gt_valid  = (const unsigned char*)d_in[8];
    const float* log_var_sys       = (const float*)d_in[9];
    const float* log_var_bar       = (const float*)d_in[10];
    const float* log_var_note      = (const float*)d_in[11];

    int Bn = in_sizes[5];   // gt_system_idx length == B

    char* ws = (char*)d_ws;
    int*   off_sys = (int*)ws;            ws += (size_t)Bn * sizeof(int);
    int*   off_bar = (int*)ws;            ws += (size_t)Bn * sizeof(int);
    float* ce_sys  = (float*)ws;          ws += (size_t)Bn * sizeof(float);
    float* ce_bar  = (float*)ws;          ws += (size_t)Bn * sizeof(float);
    int*   fl_sys  = (int*)ws;            ws += (size_t)Bn * sizeof(int);
    int*   fl_bar  = (int*)ws;            ws += (size_t)Bn * sizeof(int);

    scan_kernel<<<2, 256, 0, stream>>>(sys_counts, bar_counts, off_sys, off_bar, Bn);

    seg_ce_kernel<<<2 * Bn, SEG_BLOCK, 0, stream>>>(
        sys_logits, bar_logits, off_sys, off_bar, sys_counts, bar_counts,
        gt_system_idx, gt_bar_in_sys, gt_valid,
        ce_sys, ce_bar, fl_sys, fl_bar, Bn);

    finalize_kernel<<<1, FIN_BLOCK, 0, stream>>>(
        ce_sys, fl_sys, ce_bar, fl_bar, note_positions, gt_note_position,
        gt_valid, log_var_sys, log_var_bar, log_var_note, (float*)d_out, Bn);
}